// GNNModel_32830730011138
// MI455X (gfx1250) — compile-verified
//
#include <hip/hip_runtime.h>
#include <hip/hip_bf16.h>

// GraphSAGE 2-layer on gfx1250.
// Phase 1/3: edge gather + atomic mean-aggregate (HBM/atomic bound, L2-resident accumulators).
// Phase 2/4: fused mean-scale + dual GEMM (mean@Wl + x@Wr) + bias + ReLU using
//            V_WMMA_F32_16X16X4_F32. Each wave owns a 16x(NT*16) output strip so the
//            A fragments (mean rows / x rows) are loaded once and reused across NT
//            column tiles -> 8x less A traffic than one-tile-per-wave.

typedef __attribute__((ext_vector_type(2))) float v2f;
typedef __attribute__((ext_vector_type(8))) float v8f;

static constexpr int D  = 128;
static constexpr int N1 = 131072;
static constexpr int N2 = 8192;
static constexpr int E0 = 1048576;
static constexpr int E1 = 131072;

// ---------------------------------------------------------------------------
// Zero a contiguous float region (vectorized, grid-stride).
// ---------------------------------------------------------------------------
__global__ __launch_bounds__(256) void zero_kernel(float4* __restrict__ p, size_t n4) {
    size_t i      = (size_t)blockIdx.x * blockDim.x + threadIdx.x;
    size_t stride = (size_t)gridDim.x * blockDim.x;
    const float4 z = make_float4(0.f, 0.f, 0.f, 0.f);
    for (; i < n4; i += stride) p[i] = z;
}

// ---------------------------------------------------------------------------
// Edge scatter: one wave (32 lanes) per edge, 4 floats per lane.
// agg[dst] += xs[src]; cnt[dst] += 1.  L2 atomic units absorb the traffic
// (64 MB accumulator << 192 MB L2).
// ---------------------------------------------------------------------------
__global__ __launch_bounds__(256) void scatter_kernel(const float* __restrict__ xs,
                                                      const int* __restrict__ src,
                                                      const int* __restrict__ dst,
                                                      float* __restrict__ agg,
                                                      float* __restrict__ cnt,
                                                      int nedges) {
    long long t = (long long)blockIdx.x * blockDim.x + threadIdx.x;
    int e = (int)(t >> 5);          // wave-aligned: 32 consecutive threads share an edge
    if (e >= nedges) return;
    int q = ((int)t & 31) << 2;     // feature offset: lane*4
    int s = src[e];
    int d = dst[e];
    const float4 v = *(const float4*)(xs + (size_t)s * D + q);
    float* a = agg + (size_t)d * D + q;
    atomicAdd(a + 0, v.x);
    atomicAdd(a + 1, v.y);
    atomicAdd(a + 2, v.z);
    atomicAdd(a + 3, v.w);
    if (q == 0) atomicAdd(cnt + d, 1.0f);
}

// ---------------------------------------------------------------------------
// Fused SAGE dense stage: out = relu( (agg/max(cnt,1)) @ Wl + bl + xtgt @ Wr )
// One wave per 16-row x (NT*16)-col output strip. K=128 swept in steps of 4;
// per step: load A fragments once, then 2 accumulating WMMAs per column tile.
//
// VGPR layouts per CDNA5 ISA 7.12.2 (32-bit operands, wave32):
//   A 16x4 : lane L<16 -> row M=L, K=(0,1) in (v0,v1); lane 16+L -> K=(2,3).
//   B 4x16 : lane L<16 -> col N=L, K=(0,1) in (v0,v1); lane 16+L -> K=(2,3).
//   C/D    : vgpr v, lane L<16 -> (M=v, N=L); lane 16+L -> (M=v+8, N=L).
// ---------------------------------------------------------------------------
template <int NT>   // column tiles per wave; NT in {1,2,4,8}
__global__ __launch_bounds__(128) void sage_wmma_kernel(const float* __restrict__ agg,
                                                        const float* __restrict__ cnt,
                                                        const float* __restrict__ xtgt,
                                                        const float* __restrict__ Wl,
                                                        const float* __restrict__ bl,
                                                        const float* __restrict__ Wr,
                                                        float* __restrict__ out,
                                                        int n_tgt) {
    constexpr int STRIPS_PER_ROWBLK = (D / 16) / NT;  // column strips per 16-row block
    const int wave  = threadIdx.x >> 5;
    const int lane  = threadIdx.x & 31;
    const int strip = blockIdx.x * 4 + wave;          // 4 waves/block, 1 strip/wave
    const int m0     = (strip / STRIPS_PER_ROWBLK) * 16;
    const int n_base = (strip % STRIPS_PER_ROWBLK) * (NT * 16);
    if (m0 >= n_tgt) return;                          // wave-uniform (never taken; grids exact)

    const int r  = lane & 15;                         // A row / B col / C col within tile
    const int hi = lane >> 4;                         // half-wave select
    const int kk = hi * 2;                            // K pair handled by this half

    const float* aggRow = agg  + (size_t)(m0 + r) * D;
    const float* xRow   = xtgt + (size_t)(m0 + r) * D;
    const float  inv    = 1.0f / fmaxf(cnt[m0 + r], 1.0f);

    v8f acc[NT];
#pragma unroll
    for (int t = 0; t < NT; ++t) acc[t] = (v8f){};

    for (int k0 = 0; k0 < D; k0 += 4) {
        v2f aM = *(const v2f*)(aggRow + k0 + kk);
        aM *= inv;                                    // mean = agg / max(cnt,1), on the fly
        v2f aX = *(const v2f*)(xRow + k0 + kk);

        const float* wlRow = Wl + (size_t)(k0 + kk) * D + r;   // rows k0+kk, k0+kk+1
        const float* wrRow = Wr + (size_t)(k0 + kk) * D + r;
#pragma unroll
        for (int t = 0; t < NT; ++t) {
            const int n0 = n_base + t * 16;
            v2f bL, bR;
            bL.x = wlRow[n0];
            bL.y = wlRow[(size_t)D + n0];
            bR.x = wrRow[n0];
            bR.y = wrRow[(size_t)D + n0];
            acc[t] = __builtin_amdgcn_wmma_f32_16x16x4_f32(false, aM, false, bL,
                                                           (short)0, acc[t], false, false);
            acc[t] = __builtin_amdgcn_wmma_f32_16x16x4_f32(false, aX, false, bR,
                                                           (short)0, acc[t], false, false);
        }
    }

#pragma unroll
    for (int t = 0; t < NT; ++t) {
        const int n0   = n_base + t * 16;
        const float bias = bl[n0 + r];
#pragma unroll
        for (int v = 0; v < 8; ++v) {
            float val = fmaxf(acc[t][v] + bias, 0.0f);
            out[(size_t)(m0 + hi * 8 + v) * D + n0 + r] = val;
        }
    }
}

// ---------------------------------------------------------------------------
// Launch: zero accumulators -> scatter0 -> sage0 (h1) -> scatter1 -> sage1 (out)
// ---------------------------------------------------------------------------
extern "C" void kernel_launch(void* const* d_in, const int* in_sizes, int n_in,
                              void* d_out, int out_size, void* d_ws, size_t ws_size,
                              hipStream_t stream) {
    const float* x   = (const float*)d_in[0];
    const float* Wl0 = (const float*)d_in[1];
    const float* bl0 = (const float*)d_in[2];
    const float* Wr0 = (const float*)d_in[3];
    const float* Wl1 = (const float*)d_in[4];
    const float* bl1 = (const float*)d_in[5];
    const float* Wr1 = (const float*)d_in[6];
    const int*   src0 = (const int*)d_in[7];
    const int*   dst0 = (const int*)d_in[8];
    const int*   src1 = (const int*)d_in[9];
    const int*   dst1 = (const int*)d_in[10];
    float*       out  = (float*)d_out;

    // Workspace layout: [agg0 | cnt0 | agg1 | cnt1] (zeroed) then [h1].
    float* ws   = (float*)d_ws;
    float* agg0 = ws;                                  // N1*D
    float* cnt0 = agg0 + (size_t)N1 * D;               // N1
    float* agg1 = cnt0 + N1;                           // N2*D
    float* cnt1 = agg1 + (size_t)N2 * D;               // N2
    float* h1   = cnt1 + N2;                           // N1*D

    const size_t zn = (size_t)N1 * D + N1 + (size_t)N2 * D + N2;  // divisible by 4
    zero_kernel<<<4096, 256, 0, stream>>>((float4*)ws, zn / 4);

    // Layer 0: N0 sources -> N1 targets.  NT=8: one wave per 16x128 output strip.
    {
        int threads = E0 * 32;
        scatter_kernel<<<(threads + 255) / 256, 256, 0, stream>>>(x, src0, dst0, agg0, cnt0, E0);
        int strips = (N1 / 16) * 1;                    // 8192 waves
        sage_wmma_kernel<8><<<strips / 4, 128, 0, stream>>>(agg0, cnt0, x, Wl0, bl0, Wr0, h1, N1);
    }

    // Layer 1: N1 sources (h1) -> N2 targets.  NT=2 for more waves (small layer).
    {
        int threads = E1 * 32;
        scatter_kernel<<<(threads + 255) / 256, 256, 0, stream>>>(h1, src1, dst1, agg1, cnt1, E1);
        int strips = (N2 / 16) * 4;                    // 2048 waves
        sage_wmma_kernel<2><<<strips / 4, 128, 0, stream>>>(agg1, cnt1, h1, Wl1, bl1, Wr1, out, N2);
    }
}